// RepPhiModel_78305843740909
// MI455X (gfx1250) — compile-verified
//
#include <hip/hip_runtime.h>
#include <hip/hip_bf16.h>

// dims (fixed by the reference)
#define U_   2048
#define T_   200
#define S_   10
#define K_   20
#define DE_  32
#define DU_  32
#define H_   16
#define SK_  200          // S*K real feature columns
// Wx row offsets: emb [0,32), q [32,42), zero 42, f [43,243), zero [243,263), ctime 263

typedef __attribute__((ext_vector_type(2))) float v2f;
typedef __attribute__((ext_vector_type(8))) float v8f;

static __device__ __forceinline__ v8f wmma4(v2f a, v2f b, v8f c) {
  return __builtin_amdgcn_wmma_f32_16x16x4_f32(false, a, false, b, (short)0, c,
                                               false, false);
}

// ---------------------------------------------------------------------------
// Kernel A: E2[v][h] = emb_table[v][:] @ Wx[0:32][:,h]  via WMMA.
// 16 rows per wave, 8 chunks of K=4. Tail rows: A-loads clamped to the last
// real row; stores land in padded scratch rows we own. Surplus waves retire
// with a uniform branch (EXEC stays all-ones for every WMMA).
// ---------------------------------------------------------------------------
__global__ void emb_proj_kernel(const float* __restrict__ emb,
                                const float* __restrict__ Wx,
                                float* __restrict__ E2, int nrows, int ntiles) {
  const int wave = threadIdx.x >> 5;
  const int lane = threadIdx.x & 31;
  const int ln   = lane & 15;
  const int half = lane >> 4;
  const int tile = blockIdx.x * 8 + wave;
  if (tile >= ntiles) return;                       // uniform per wave

  const long g0 = (long)tile * 16;
  long vr = g0 + ln;                                // A-frag source row
  if (vr > nrows - 1) vr = nrows - 1;               // clamp tail loads
  const float* er = emb + vr * DE_;

  v8f c;
#pragma unroll
  for (int j = 0; j < 8; ++j) c[j] = 0.f;

#pragma unroll
  for (int kc = 0; kc < 8; ++kc) {
    const int k = kc * 4 + half * 2;
    v2f a, b;
    a.x = er[k];
    a.y = er[k + 1];
    b.x = Wx[(k)     * H_ + ln];
    b.y = Wx[(k + 1) * H_ + ln];
    c = wmma4(a, b, c);
  }

#pragma unroll
  for (int j = 0; j < 8; ++j)
    E2[(g0 + j + 8 * half) * H_ + ln] = c[j];       // padded region is ours
}

// ---------------------------------------------------------------------------
// Kernel B: pre[g][h] = E2[docid[g]][h] + b[h]
//                     + [q(10) | ctime | 0] @ [Wx[32:42] | Wx[263] | 0]
//                     + f[g][:200] @ Wx[43:243]
// One wave per 16-row tile; 53 chained v_wmma_f32_16x16x4_f32, no divergence.
// ---------------------------------------------------------------------------
__global__ void pre_gemm_kernel(const float* __restrict__ q,      // [U*T, 10]
                                const float* __restrict__ f,      // [U*T, 200]
                                const int*   __restrict__ docid,  // [U*T]
                                const float* __restrict__ ct,     // [U*T]
                                const float* __restrict__ Wx,     // [264, 16]
                                const float* __restrict__ b_rnn,  // [16]
                                const float* __restrict__ E2,     // padded
                                float* __restrict__ pre)          // [U*T, 16]
{
  const int wave = threadIdx.x >> 5;
  const int lane = threadIdx.x & 31;
  const int ln   = lane & 15;      // A row m == B/C/D col n
  const int half = lane >> 4;
  const long g0  = ((long)blockIdx.x * 8 + wave) * 16;

  // --- accumulator init: E2 gather + bias (C/D layout) ---
  v8f c;
  const float bn = b_rnn[ln];
#pragma unroll
  for (int j = 0; j < 8; ++j) {
    long g = g0 + j + 8 * half;
    c[j] = E2[(long)docid[g] * H_ + ln] + bn;
  }

  // --- q chunks 0,1: K = 0..7, fully in-range, unconditional loads ---
  const float* qrow = q + (g0 + ln) * S_;
#pragma unroll
  for (int kc = 0; kc < 2; ++kc) {
    const int k = kc * 4 + half * 2;
    v2f a, b;
    a.x = qrow[k];
    a.y = qrow[k + 1];
    b.x = Wx[(32 + k)     * H_ + ln];
    b.y = Wx[(32 + k + 1) * H_ + ln];
    c = wmma4(a, b, c);
  }

  // --- chunk 2: K = {q8, q9, ctime, 0}; branchless value-selects ---
  {
    float q8 = qrow[8], q9 = qrow[9];
    float cg = ct[g0 + ln];
    float w8 = Wx[(32 + 8) * H_ + ln];
    float w9 = Wx[(32 + 9) * H_ + ln];
    float wc = Wx[263 * H_ + ln];
    v2f a, b;
    a.x = half ? cg  : q8;
    a.y = half ? 0.f : q9;
    b.x = half ? wc  : w8;
    b.y = half ? 0.f : w9;
    c = wmma4(a, b, c);
  }

  // --- features: K = 200, 5 groups x 10 unrolled chunks of K=4 ---
  const float* frow = f  + (g0 + ln) * SK_ + half * 2;   // 8B/lane/iter
  const float* wrow = Wx + (43 + half * 2) * H_ + ln;    // cache-hot Wx
  for (int kg = 0; kg < 5; ++kg) {
    __builtin_prefetch(frow + kg * 40 + 80, 0, 0);
#pragma unroll
    for (int ki = 0; ki < 10; ++ki) {
      const int kc = kg * 10 + ki;
      v2f a, b;
      a.x = frow[kc * 4 + 0];
      a.y = frow[kc * 4 + 1];
      b.x = wrow[(kc * 4) * H_];
      b.y = wrow[(kc * 4) * H_ + H_];
      c = wmma4(a, b, c);
    }
  }

  // --- store D tile (64B per half-wave per row) ---
#pragma unroll
  for (int j = 0; j < 8; ++j) {
    long g = g0 + j + 8 * half;
    pre[g * H_ + ln] = c[j];
  }
}

// ---------------------------------------------------------------------------
// Kernel C: masked SimpleRNN over T (serial, latency-bound) + LeakyReLU + Dense
// 2 users per wave: lane = (usub, hidden j); Wh column in registers,
// h broadcast via cross-lane shuffles within each 16-lane user group.
// ---------------------------------------------------------------------------
__global__ void rnn_dense_kernel(const float* __restrict__ pre,     // [U*T, 16]
                                 const int*   __restrict__ docid,   // [U*T]
                                 const float* __restrict__ Wh,      // [16,16]
                                 const float* __restrict__ Wd,      // [16,32]
                                 const float* __restrict__ b_dense, // [32]
                                 float* __restrict__ out)           // [U,32]
{
  const int lane = threadIdx.x & 31;
  const int usub = lane >> 4;
  const int j    = lane & 15;
  const int wave = threadIdx.x >> 5;
  const int u    = (blockIdx.x * 8 + wave) * 2 + usub;

  float whcol[H_];
#pragma unroll
  for (int nn = 0; nn < H_; ++nn) whcol[nn] = Wh[nn * H_ + j];

  float h = 0.f;
  const float* prow = pre + (long)u * T_ * H_ + j;
  const int*   drow = docid + (long)u * T_;
  for (int t = 0; t < T_; ++t) {
    float acc = prow[t * H_];
#pragma unroll
    for (int nn = 0; nn < H_; ++nn) {
      float hn = __shfl(h, usub * 16 + nn, 32);
      acc += hn * whcol[nn];
    }
    float hnew = tanhf(acc);
    h = (drow[t] != 0) ? hnew : h;   // mask_zero: skipped steps keep state
  }

  float a = (h >= 0.f) ? h : 0.3f * h;  // LeakyReLU(0.3)
  float acc0 = b_dense[j], acc1 = b_dense[j + 16];
#pragma unroll
  for (int nn = 0; nn < H_; ++nn) {
    float an = __shfl(a, usub * 16 + nn, 32);
    acc0 += an * Wd[nn * DU_ + j];
    acc1 += an * Wd[nn * DU_ + j + 16];
  }
  out[(long)u * DU_ + j]      = acc0;
  out[(long)u * DU_ + j + 16] = acc1;
}

// ---------------------------------------------------------------------------
extern "C" void kernel_launch(void* const* d_in, const int* in_sizes, int n_in,
                              void* d_out, int out_size, void* d_ws, size_t ws_size,
                              hipStream_t stream) {
  const float* q     = (const float*)d_in[0];
  const float* f     = (const float*)d_in[1];
  const int*   docid = (const int*)  d_in[2];
  const float* ct    = (const float*)d_in[3];
  const float* emb   = (const float*)d_in[4];
  const float* Wx    = (const float*)d_in[5];
  const float* Wh    = (const float*)d_in[6];
  const float* brnn  = (const float*)d_in[7];
  const float* Wd    = (const float*)d_in[8];
  const float* bd    = (const float*)d_in[9];
  float* out = (float*)d_out;

  const int VP1    = in_sizes[4] / DE_;           // 100001 rows in emb table
  const int ntiles = (VP1 + 15) / 16;             // 6251 row-tiles
  char* ws = (char*)d_ws;
  float* E2  = (float*)ws;                        // [ntiles*16, 16] f32 (padded)
  size_t e2b = (((size_t)ntiles * 16 * H_ * sizeof(float)) + 255) & ~(size_t)255;
  float* pre = (float*)(ws + e2b);                // [U*T, 16] f32

  emb_proj_kernel<<<(ntiles + 7) / 8, 256, 0, stream>>>(emb, Wx, E2, VP1, ntiles);

  const int tiles = (U_ * T_) / 16;               // 25600, 8 waves per block
  pre_gemm_kernel<<<tiles / 8, 256, 0, stream>>>(q, f, docid, ct, Wx, brnn, E2, pre);

  rnn_dense_kernel<<<U_ / 16, 256, 0, stream>>>(pre, docid, Wh, Wd, bd, out);
}